// GraphRegressionModel_41162966564945
// MI455X (gfx1250) — compile-verified
//
#include <hip/hip_runtime.h>

#define HD 64  // hidden dim

typedef __attribute__((ext_vector_type(2))) float v2f;
typedef __attribute__((ext_vector_type(8))) float v8f;

__device__ __forceinline__ void atomAddF(float* p, float v) {
    unsafeAtomicAdd(p, v);   // lowers to global_atomic_add_f32 (L2 RMW), not CAS loop
}

// ---------------- utility ----------------
__global__ void zero_f32(float* __restrict__ p, long n) {
    long i = (long)blockIdx.x * blockDim.x + threadIdx.x;
    if (i < n) p[i] = 0.0f;
}

// ---------------- layer 1 fast path ----------------
// x0 = ones(N,1)  =>  h = x0 @ W1 has identical rows == W1.
// scatter(h[src]*w) = W1 * degw[dst], so layer1 reduces to scalar degree scatter.
__global__ void edge_degw(const int* __restrict__ dst, const float* __restrict__ w,
                          float* __restrict__ degw, int E_) {
    int e = blockIdx.x * blockDim.x + threadIdx.x;
    if (e < E_) atomAddF(&degw[dst[e]], w[e]);
}

__global__ void layer1(const float* __restrict__ degw, const float* __restrict__ W1,
                       const float* __restrict__ b1, float* __restrict__ x, int N_) {
    long i = (long)blockIdx.x * blockDim.x + threadIdx.x;
    if (i >= (long)N_ * HD) return;
    int n = (int)(i >> 6), j = (int)(i & 63);
    float v = fmaf(W1[j], degw[n], b1[j]);
    x[i] = v > 0.0f ? v : 0.0f;
}

// ---------------- h = x @ W  via V_WMMA_F32_16X16X4_F32 ----------------
// Block = 256 threads = 8 waves. Block covers 32 rows x 64 cols.
// W (16 KB) staged into LDS via GLOBAL_LOAD_ASYNC_TO_LDS_B128 (ASYNCcnt DMA path):
// each thread moves its 64 B of W straight to LDS, no VGPR round-trip.
__global__ void gemm_wmma(const float* __restrict__ x, const float* __restrict__ W,
                          float* __restrict__ h, int N_) {
    __shared__ float sW[HD * HD];   // 16 KB of W in LDS
    int tid = threadIdx.x;

    {
        // thread t owns bytes [t*64, t*64+64) of W, copied with 4 async b128s
        const float* gsrc = W + tid * 16;
        unsigned lds_addr = (unsigned)(uintptr_t)(&sW[tid * 16]);
        asm volatile(
            "global_load_async_to_lds_b128 %0, %1, off\n\t"
            "global_load_async_to_lds_b128 %0, %1, off offset:16\n\t"
            "global_load_async_to_lds_b128 %0, %1, off offset:32\n\t"
            "global_load_async_to_lds_b128 %0, %1, off offset:48\n\t"
            "s_wait_asynccnt 0"
            :: "v"(lds_addr), "v"(gsrc) : "memory");
    }
    __syncthreads();

    int wave = tid >> 5;
    int lane = tid & 31;
    int rt = wave >> 2;
    int ct = wave & 3;
    int row0 = blockIdx.x * 32 + rt * 16;
    int col0 = ct * 16;

    int m    = lane & 15;           // M (for A) / N (for B) position within tile
    int koff = (lane >> 4) * 2;     // lanes 16-31 carry K+2, K+3

    int arow = row0 + m;
    if (arow >= N_) arow = N_ - 1;  // branchless clamp: EXEC stays all-ones for WMMA
    const float* xr = x + (long)arow * HD;

    v8f c = {};
#pragma unroll
    for (int k = 0; k < HD; k += 4) {
        v2f a, b;
        // A 16x4 fp32: VGPR0 = K+koff, VGPR1 = K+koff+1 (ISA 7.12.2 layout)
        a.x = xr[k + koff];
        a.y = xr[k + koff + 1];
        // B 4x16 fp32: rows K striped like C/D (lanes 0-15: K, K+1; lanes 16-31: K+2, K+3)
        b.x = sW[(k + koff) * HD + col0 + m];
        b.y = sW[(k + koff + 1) * HD + col0 + m];
        c = __builtin_amdgcn_wmma_f32_16x16x4_f32(
                /*neg_a=*/false, a, /*neg_b=*/false, b,
                /*c_mod=*/(short)0, c, /*reuse_a=*/false, /*reuse_b=*/false);
    }

    // D 16x16 fp32: VGPR r -> M = r (lanes 0-15) or M = 8 + r (lanes 16-31)
    int orow_base = row0 + ((lane >> 4) << 3);
    int ocol = col0 + m;
#pragma unroll
    for (int r = 0; r < 8; ++r) {
        int orow = orow_base + r;
        if (orow < N_) h[(long)orow * HD + ocol] = c[r];
    }
}

// ---------------- weighted edge scatter: out[dst] += h[src] * w ----------------
// One thread handles 4 contiguous channels of one edge (float4 gather, 4 atomics).
__global__ void edge_scatter(const int* __restrict__ src, const int* __restrict__ dst,
                             const float* __restrict__ w, const float* __restrict__ h,
                             float* __restrict__ out, int E_) {
    long i = (long)blockIdx.x * blockDim.x + threadIdx.x;
    long e = i >> 4;
    if (e >= E_) return;
    int j = (int)(i & 15) * 4;
    int s = src[e], d = dst[e];
    float we = w[e];
    float4 hv = *(const float4*)(h + (long)s * HD + j);
    float* op = out + (long)d * HD + j;
    atomAddF(op + 0, hv.x * we);
    atomAddF(op + 1, hv.y * we);
    atomAddF(op + 2, hv.z * we);
    atomAddF(op + 3, hv.w * we);
}

__global__ void bias_relu(float* __restrict__ x, const float* __restrict__ b, int N_) {
    long i = (long)blockIdx.x * blockDim.x + threadIdx.x;
    if (i >= (long)N_ * HD) return;
    float v = x[i] + b[i & 63];
    x[i] = v > 0.0f ? v : 0.0f;
}

// ---------------- pooling + head ----------------
__global__ void pool_sum(const float* __restrict__ x, const int* __restrict__ batch,
                         float* __restrict__ sums, int N_) {
    long i = (long)blockIdx.x * blockDim.x + threadIdx.x;
    if (i >= (long)N_ * HD) return;
    int n = (int)(i >> 6), j = (int)(i & 63);
    atomAddF(&sums[(long)batch[n] * HD + j], x[i]);
}

__global__ void pool_cnt(const int* __restrict__ batch, float* __restrict__ counts, int N_) {
    int n = blockIdx.x * blockDim.x + threadIdx.x;
    if (n < N_) atomAddF(&counts[batch[n]], 1.0f);
}

__global__ void final_lin(const float* __restrict__ sums, const float* __restrict__ counts,
                          const float* __restrict__ lin_w, const float* __restrict__ lin_b,
                          float* __restrict__ out, int G) {
    int g = blockIdx.x * blockDim.x + threadIdx.x;
    if (g >= G) return;
    float cnt = counts[g];
    if (cnt < 1.0f) cnt = 1.0f;
    float acc = 0.0f;
#pragma unroll
    for (int j = 0; j < HD; ++j) acc += sums[g * HD + j] * lin_w[j];
    out[g] = acc / cnt + lin_b[0];
}

// ---------------- driver ----------------
static inline unsigned cdiv(long a, long b) { return (unsigned)((a + b - 1) / b); }

extern "C" void kernel_launch(void* const* d_in, const int* in_sizes, int n_in,
                              void* d_out, int out_size, void* d_ws, size_t ws_size,
                              hipStream_t stream) {
    const int*   edge_index  = (const int*)d_in[0];   // (2, E)
    const float* edge_weight = (const float*)d_in[1]; // (E,)
    const int*   batch       = (const int*)d_in[2];   // (N,)
    const float* W1   = (const float*)d_in[4];
    const float* b1   = (const float*)d_in[5];
    const float* Ws[4] = { (const float*)d_in[6],  (const float*)d_in[8],
                           (const float*)d_in[10], (const float*)d_in[12] };
    const float* bs[4] = { (const float*)d_in[7],  (const float*)d_in[9],
                           (const float*)d_in[11], (const float*)d_in[13] };
    const float* lin_w = (const float*)d_in[14];
    const float* lin_b = (const float*)d_in[15];
    float* out = (float*)d_out;

    const int E_ = in_sizes[1];
    const int N_ = in_sizes[2];
    const int G  = out_size;
    const int* src = edge_index;
    const int* dst = edge_index + E_;

    // workspace layout (~52 MB)
    float* xbuf   = (float*)d_ws;                 // N*H
    float* hbuf   = xbuf + (size_t)N_ * HD;       // N*H
    float* degw   = hbuf + (size_t)N_ * HD;       // N
    float* sums   = degw + N_;                    // G*H
    float* counts = sums + (size_t)G * HD;        // G

    const long NH = (long)N_ * HD;

    // ---- layer 1 (degree shortcut) ----
    zero_f32<<<cdiv(N_, 256), 256, 0, stream>>>(degw, N_);
    edge_degw<<<cdiv(E_, 256), 256, 0, stream>>>(dst, edge_weight, degw, E_);
    layer1<<<cdiv(NH, 256), 256, 0, stream>>>(degw, W1, b1, xbuf, N_);

    // ---- layers 2..5 ----
    for (int l = 0; l < 4; ++l) {
        gemm_wmma<<<cdiv(N_, 32), 256, 0, stream>>>(xbuf, Ws[l], hbuf, N_);
        zero_f32<<<cdiv(NH, 256), 256, 0, stream>>>(xbuf, NH);
        edge_scatter<<<cdiv((long)E_ * 16, 256), 256, 0, stream>>>(
            src, dst, edge_weight, hbuf, xbuf, E_);
        bias_relu<<<cdiv(NH, 256), 256, 0, stream>>>(xbuf, bs[l], N_);
    }

    // ---- global mean pool + linear head ----
    zero_f32<<<cdiv((long)G * HD + G, 256), 256, 0, stream>>>(sums, (long)G * HD + G);
    pool_sum<<<cdiv(NH, 256), 256, 0, stream>>>(xbuf, batch, sums, N_);
    pool_cnt<<<cdiv(N_, 256), 256, 0, stream>>>(batch, counts, N_);
    final_lin<<<1, 64, 0, stream>>>(sums, counts, lin_w, lin_b, out, G);
}